// IWT_20315195310496
// MI455X (gfx1250) — compile-verified
//
#include <hip/hip_runtime.h>
#include <cstdint>

// Inverse 2x2 Haar wavelet synthesis, MI455X (gfx1250).
// x:   (4, 256, 256, 256) f32, channel dim = 64 groups of [a,h,v,d] planes
// out: (4, 64, 512, 512)  f32
//
// Bandwidth-bound (512 MiB traffic, ~23 us floor @ 23.3 TB/s).
// Data path: GLOBAL_LOAD_ASYNC_TO_LDS_B128 (ASYNCcnt, non-temporal) ->
// ds_load_b128 -> butterfly in VALU -> non-temporal global_store_b128.
// Both streams are touched exactly once (512 MiB >> 192 MB L2), so both
// directions carry NT temporal hints.

typedef float v4f __attribute__((ext_vector_type(4)));

__global__ __launch_bounds__(256) void iwt_haar_gfx1250(const float* __restrict__ x,
                                                        float* __restrict__ out) {
  // 16 KiB LDS: plane-major [plane][tid] slots of 16 B each.
  __shared__ v4f lds_buf[4 * 256];

  const int tid = threadIdx.x;
  const int t   = blockIdx.x * 256 + tid;

  const int w4 = t & 63;          // 4-float chunk along W2 (0..63)
  const int h  = (t >> 6) & 255;  // 0..255
  const int c  = (t >> 14) & 63;  // 0..63
  const int b  = t >> 20;         // 0..3

  // Input: plane p lives at channel c*4+p; plane stride = 256*256 floats.
  const float* gsrc = x + (((b * 256 + c * 4) * 256 + h) * 256 + w4 * 4);

  // Issue 4 async 16B copies global->LDS (one per wavelet plane), NT hint.
  // VDST operand = LDS byte address (low 32 bits of the generic pointer).
#pragma unroll
  for (int p = 0; p < 4; ++p) {
    unsigned loff    = (unsigned)(uintptr_t)(&lds_buf[p * 256 + tid]);
    const float* ga  = gsrc + p * (256 * 256);
    asm volatile("global_load_async_to_lds_b128 %0, %1, off th:TH_LOAD_NT"
                 :: "v"(loff), "v"(ga)
                 : "memory");
  }
  // Wave-private staging: each lane reads only slots its own wave wrote,
  // so waiting on this wave's ASYNCcnt is sufficient (no barrier needed).
  asm volatile("s_wait_asynccnt 0" ::: "memory");

  v4f a  = lds_buf[0 * 256 + tid];
  v4f hh = lds_buf[1 * 256 + tid];
  v4f vv = lds_buf[2 * 256 + tid];
  v4f d  = lds_buf[3 * 256 + tid];

  // Factored butterfly: 8 add/sub + 4 mul per float4.
  v4f t0 = a + vv;
  v4f t1 = hh + d;
  v4f t2 = a - vv;
  v4f t3 = hh - d;
  v4f p00 = (t0 + t1) * 0.5f;  // out[2h  ][2w  ]
  v4f p01 = (t0 - t1) * 0.5f;  // out[2h  ][2w+1]
  v4f p10 = (t2 + t3) * 0.5f;  // out[2h+1][2w  ]
  v4f p11 = (t2 - t3) * 0.5f;  // out[2h+1][2w+1]

  // Interleave pairs along the output row.
  v4f top0 = __builtin_shufflevector(p00, p01, 0, 4, 1, 5);
  v4f top1 = __builtin_shufflevector(p00, p01, 2, 6, 3, 7);
  v4f bot0 = __builtin_shufflevector(p10, p11, 0, 4, 1, 5);
  v4f bot1 = __builtin_shufflevector(p10, p11, 2, 6, 3, 7);

  // Output: (b*64+c) channel of 512x512; rows 2h and 2h+1, cols 8*w4..8*w4+7.
  float* orow = out + (((b * 64 + c) * 512 + 2 * h) * 512 + w4 * 8);

  // Write-once output: non-temporal stores.
  __builtin_nontemporal_store(top0, (v4f*)(orow));
  __builtin_nontemporal_store(top1, (v4f*)(orow + 4));
  __builtin_nontemporal_store(bot0, (v4f*)(orow + 512));
  __builtin_nontemporal_store(bot1, (v4f*)(orow + 516));
}

extern "C" void kernel_launch(void* const* d_in, const int* in_sizes, int n_in,
                              void* d_out, int out_size, void* d_ws, size_t ws_size,
                              hipStream_t stream) {
  const float* x = (const float*)d_in[0];
  float* out     = (float*)d_out;

  // Total threads = 4 * 64 * 256 * (256/4) = 4,194,304 -> 16384 blocks of 256.
  dim3 grid(16384), block(256);
  hipLaunchKernelGGL(iwt_haar_gfx1250, grid, block, 0, stream, x, out);
}